// GraphConvolutionBatch_26774826123628
// MI455X (gfx1250) — compile-verified
//
#include <hip/hip_runtime.h>
#include <hip/hip_bf16.h>

// ---------------------------------------------------------------------------
// GraphConvolutionBatch: Hagg = d * stencil9( d * relu( BN( H @ W ) ) )
//   H: [16, 65536, 64] f32, W: [64,64] f32, gamma/beta: [64] f32
// Memory-bound (~1.07 GB min traffic @ 23.3 TB/s). Three kernels:
//   1) GEMM via v_wmma_f32_16x16x4_f32 + deterministic per-block stats.
//      W staged in LDS in K-paired layout so every B fragment is one
//      8B-aligned ds_load_b64 straight into an even VGPR pair (no remat movs).
//   2) stats finalize (double accumulation) -> folded BN scale/shift
//   3) fused BN+ReLU+9-point stencil with analytic degree weights
// ---------------------------------------------------------------------------

typedef float v2f __attribute__((ext_vector_type(2)));
typedef float v8f __attribute__((ext_vector_type(8)));

#define GB_BATCH   16
#define GB_R       256
#define GB_C       256
#define GB_N       (GB_R * GB_C)            // 65536
#define GB_TOTROWS (GB_BATCH * GB_N)        // 1048576
#define GB_F       64
#define GB_ROWS_PER_BLOCK 128
#define GB_GEMM_BLOCKS (GB_TOTROWS / GB_ROWS_PER_BLOCK)  // 8192
#define GB_EPS_BN  1e-5f
#define GB_EPS_DEG 1e-5f

// -------------------- Kernel 1: H @ W with WMMA + block stats ---------------
// Block = 256 threads = 8 waves. Wave w handles rows [blk*128 + w*16, +16),
// producing a 16x64 tile of Ht via 4 accumulators of V_WMMA_F32_16X16X4_F32.
__global__ __launch_bounds__(256)
void gb_gemm_stats_kernel(const float* __restrict__ H,
                          const float* __restrict__ W,
                          float* __restrict__ Ht,
                          float* __restrict__ partials) {
    // K-paired W image: Wp[(kk>>1)*128 + n*2 + (kk&1)] = W[kk][n]
    // => B fragment {W[kk][n], W[kk+1][n]} is contiguous & 8B aligned.
    __shared__ float Wp[GB_F * GB_F];      // 16 KB
    __shared__ float red[2 * 16 * GB_F];   // [sum|sq][wave*2+half][feat], 8 KB

    const int t = threadIdx.x;

    for (int i = t; i < GB_F * GB_F; i += 256) {
        const int kk = i >> 6;   // input-dim index (row of W)
        const int n  = i & 63;   // output-dim index (col of W)
        Wp[((kk >> 1) << 7) + (n << 1) + (kk & 1)] = W[i];
    }
    __syncthreads();

    const int wave = t >> 5;
    const int lane = t & 31;
    const int l16  = lane & 15;
    const int lh   = lane >> 4;           // half-wave select

    const size_t rowBase = (size_t)blockIdx.x * GB_ROWS_PER_BLOCK + (size_t)wave * 16;
    const float* hrow = H + (rowBase + (size_t)l16) * GB_F;

    v8f acc0 = {}, acc1 = {}, acc2 = {}, acc3 = {};

    #pragma unroll
    for (int k = 0; k < GB_F; k += 4) {
        // A fragment: lanes 0-15 carry K = k..k+1, lanes 16-31 carry K = k+2..k+3
        v2f A = *(const v2f*)(hrow + k + 2 * lh);
        // B fragments: one ds_load_b64 each, {row kk, row kk+1} at col nb*16+l16
        const float* wb = Wp + (((k >> 1) + lh) << 7) + (l16 << 1);
        v2f B0 = *(const v2f*)(wb + 0);
        v2f B1 = *(const v2f*)(wb + 32);
        v2f B2 = *(const v2f*)(wb + 64);
        v2f B3 = *(const v2f*)(wb + 96);
        acc0 = __builtin_amdgcn_wmma_f32_16x16x4_f32(false, A, false, B0, (short)0, acc0, false, false);
        acc1 = __builtin_amdgcn_wmma_f32_16x16x4_f32(false, A, false, B1, (short)0, acc1, false, false);
        acc2 = __builtin_amdgcn_wmma_f32_16x16x4_f32(false, A, false, B2, (short)0, acc2, false, false);
        acc3 = __builtin_amdgcn_wmma_f32_16x16x4_f32(false, A, false, B3, (short)0, acc3, false, false);
    }

    // Store Ht tile. C/D layout: VGPR j -> row (j + 8*lh), col (nb*16 + l16).
    float* orow = Ht + (rowBase + (size_t)(8 * lh)) * GB_F + l16;
    #pragma unroll
    for (int j = 0; j < 8; ++j) {
        orow[(size_t)j * GB_F + 0]  = acc0[j];
        orow[(size_t)j * GB_F + 16] = acc1[j];
        orow[(size_t)j * GB_F + 32] = acc2[j];
        orow[(size_t)j * GB_F + 48] = acc3[j];
    }

    // Per-lane partial stats over the 8 rows this lane holds, per feature.
    float s0 = 0.f, s1 = 0.f, s2 = 0.f, s3 = 0.f;
    float q0 = 0.f, q1 = 0.f, q2 = 0.f, q3 = 0.f;
    #pragma unroll
    for (int j = 0; j < 8; ++j) {
        float v;
        v = acc0[j]; s0 += v; q0 += v * v;
        v = acc1[j]; s1 += v; q1 += v * v;
        v = acc2[j]; s2 += v; q2 += v * v;
        v = acc3[j]; s3 += v; q3 += v * v;
    }
    const int slot = (wave * 2 + lh) * GB_F + l16;   // [0,1024)
    red[slot + 0]         = s0;  red[slot + 16]         = s1;
    red[slot + 32]        = s2;  red[slot + 48]         = s3;
    red[1024 + slot + 0]  = q0;  red[1024 + slot + 16]  = q1;
    red[1024 + slot + 32] = q2;  red[1024 + slot + 48]  = q3;
    __syncthreads();

    // Deterministic fixed-order reduction across the 16 (wave,half) slices.
    if (t < 128) {
        const int which = t >> 6;        // 0 = sum, 1 = sumsq
        const int f     = t & 63;
        float s = 0.f;
        #pragma unroll
        for (int i = 0; i < 16; ++i)
            s += red[which * 1024 + i * GB_F + f];
        partials[(size_t)blockIdx.x * 128 + t] = s;
    }
}

// -------------------- Kernel 2: finalize BN coefficients --------------------
__global__ __launch_bounds__(128)
void gb_finalize_kernel(const float* __restrict__ partials,
                        const float* __restrict__ gamma,
                        const float* __restrict__ beta,
                        float* __restrict__ coefA,
                        float* __restrict__ coefB,
                        int nblk) {
    __shared__ double red[128];
    const int t = threadIdx.x;
    double s = 0.0;
    for (int i = 0; i < nblk; ++i)
        s += (double)partials[(size_t)i * 128 + t];
    red[t] = s;
    __syncthreads();
    if (t < 64) {
        const double inv_n = 1.0 / (double)GB_TOTROWS;
        const double mean  = red[t] * inv_n;
        const double var   = red[64 + t] * inv_n - mean * mean;
        const double istd  = 1.0 / sqrt(var + (double)GB_EPS_BN);
        const double a     = (double)gamma[t] * istd;
        coefA[t] = (float)a;
        coefB[t] = (float)((double)beta[t] - mean * a);
    }
}

// -------------------- Kernel 3: fused BN + ReLU + stencil -------------------
__device__ __forceinline__ float gb_dinv(int r, int c) {
    const float rf = (r == 0 || r == GB_R - 1) ? 2.f : 3.f;
    const float cf = (c == 0 || c == GB_C - 1) ? 2.f : 3.f;
    return 1.0f / sqrtf(rf * cf + GB_EPS_DEG);
}

__global__ __launch_bounds__(256)
void gb_stencil_kernel(const float* __restrict__ Ht,
                       const float* __restrict__ coefA,
                       const float* __restrict__ coefB,
                       float* __restrict__ out) {
    const int tid  = blockIdx.x * 256 + threadIdx.x;
    const int f4   = tid & 15;            // which float4 chunk of 64 feats
    const int node = tid >> 4;            // b*65536 + r*256 + c
    const int c    = node & (GB_C - 1);
    const int r    = (node >> 8) & (GB_R - 1);
    const int base = node & ~(GB_N - 1);  // batch base

    const float4 a  = ((const float4*)coefA)[f4];
    const float4 bb = ((const float4*)coefB)[f4];
    const float4* H4 = (const float4*)Ht;

    float4 acc = make_float4(0.f, 0.f, 0.f, 0.f);
    #pragma unroll
    for (int dr = -1; dr <= 1; ++dr) {
        const int rr = r + dr;
        if ((unsigned)rr >= (unsigned)GB_R) continue;
        #pragma unroll
        for (int dc = -1; dc <= 1; ++dc) {
            const int cc = c + dc;
            if ((unsigned)cc >= (unsigned)GB_C) continue;
            const float dj = gb_dinv(rr, cc);
            const float4 v = H4[(size_t)(base + (rr << 8) + cc) * 16 + f4];
            acc.x = fmaf(dj, fmaxf(fmaf(a.x, v.x, bb.x), 0.f), acc.x);
            acc.y = fmaf(dj, fmaxf(fmaf(a.y, v.y, bb.y), 0.f), acc.y);
            acc.z = fmaf(dj, fmaxf(fmaf(a.z, v.z, bb.z), 0.f), acc.z);
            acc.w = fmaf(dj, fmaxf(fmaf(a.w, v.w, bb.w), 0.f), acc.w);
        }
    }
    const float di = gb_dinv(r, c);
    float4 o;
    o.x = di * acc.x; o.y = di * acc.y; o.z = di * acc.z; o.w = di * acc.w;
    ((float4*)out)[(size_t)node * 16 + f4] = o;
}

// ---------------------------------------------------------------------------
extern "C" void kernel_launch(void* const* d_in, const int* in_sizes, int n_in,
                              void* d_out, int out_size, void* d_ws, size_t ws_size,
                              hipStream_t stream) {
    (void)in_sizes; (void)n_in; (void)out_size; (void)ws_size;

    const float* H     = (const float*)d_in[0];
    const float* W     = (const float*)d_in[1];
    const float* gamma = (const float*)d_in[2];
    const float* beta  = (const float*)d_in[3];
    float* out = (float*)d_out;

    // Workspace layout (floats): [0,64) coefA | [64,128) coefB | pad to 256 |
    // [256, 256 + 8192*128) partials | Ht (16*65536*64 floats), 16B aligned.
    float* ws       = (float*)d_ws;
    float* coefA    = ws;
    float* coefB    = ws + 64;
    float* partials = ws + 256;
    float* Ht       = ws + 256 + (size_t)GB_GEMM_BLOCKS * 128;

    gb_gemm_stats_kernel<<<GB_GEMM_BLOCKS, 256, 0, stream>>>(H, W, Ht, partials);
    gb_finalize_kernel<<<1, 128, 0, stream>>>(partials, gamma, beta, coefA, coefB, GB_GEMM_BLOCKS);
    gb_stencil_kernel<<<(GB_TOTROWS * 16) / 256, 256, 0, stream>>>(Ht, coefA, coefB, out);
}